// SESTKGCN_77103252897976
// MI455X (gfx1250) — compile-verified
//
#include <hip/hip_runtime.h>
#include <hip/hip_bf16.h>
#include <math.h>

#define DIM 64
#define NS  8

typedef float v2f __attribute__((ext_vector_type(2)));
typedef float v8f __attribute__((ext_vector_type(8)));

__device__ __forceinline__ float4 ldg4(const float* p) { return *(const float4*)p; }

__device__ __forceinline__ void fma4(float4& a, float w, const float4 f) {
    a.x = fmaf(w, f.x, a.x); a.y = fmaf(w, f.y, a.y);
    a.z = fmaf(w, f.z, a.z); a.w = fmaf(w, f.w, a.w);
}

__device__ __forceinline__ void softmax8(float* w) {
    float m = -1e30f;
    #pragma unroll
    for (int s = 0; s < NS; ++s) m = fmaxf(m, w[s]);
    float sum = 0.f;
    #pragma unroll
    for (int s = 0; s < NS; ++s) { w[s] = __expf(w[s] - m); sum += w[s]; }
    float inv = 1.f / sum;
    #pragma unroll
    for (int s = 0; s < NS; ++s) w[s] *= inv;
}

// ---------------------------------------------------------------------------
// Stage 1: gather + softmax-weighted aggregation. 16 lanes per batch element,
// each lane owns a float4 slice (16B gathers). Outputs user_x / item_x.
// ---------------------------------------------------------------------------
__global__ void __launch_bounds__(256) sestkgcn_agg(
    const int* __restrict__ u, const int* __restrict__ v,
    const float* __restrict__ usr_feat, const float* __restrict__ item_feat,
    const float* __restrict__ rel_feat,
    const int* __restrict__ neigh_uu, const float* __restrict__ neigh_uu_st,
    const int* __restrict__ neigh_ui, const float* __restrict__ nui_rat,
    const float* __restrict__ nui_vot, const float* __restrict__ nui_tim,
    const int* __restrict__ neigh_iu, const float* __restrict__ niu_rat,
    const float* __restrict__ niu_vot, const float* __restrict__ niu_tim,
    const int* __restrict__ neigh_ii, const int* __restrict__ neigh_ir,
    float* __restrict__ user_x, float* __restrict__ item_x, int batch)
{
    int t = blockIdx.x * 256 + threadIdx.x;
    int b = t >> 4;
    if (b >= batch) return;
    int d0 = (t & 15) * 4;

    int uu = u[b];
    int vv = v[b];

    float4 ue = ldg4(usr_feat  + (size_t)uu * DIM + d0);
    float4 ve = ldg4(item_feat + (size_t)vv * DIM + d0);

    // ---- softmax weights (redundant across the 16 lanes; cache-resident) ----
    float w_uu[NS], w_ui[NS], w_iu[NS], w_ii[NS];
    #pragma unroll
    for (int s = 0; s < NS; ++s) w_uu[s] = neigh_uu_st[uu * NS + s];
    softmax8(w_uu);
    #pragma unroll
    for (int s = 0; s < NS; ++s)
        w_ui[s] = nui_rat[uu * NS + s] * nui_vot[uu * NS + s] * nui_tim[uu * NS + s];
    softmax8(w_ui);
    #pragma unroll
    for (int s = 0; s < NS; ++s)
        w_iu[s] = niu_rat[vv * NS + s] * niu_vot[vv * NS + s] * niu_tim[vv * NS + s];
    softmax8(w_iu);

    // KGCN attention: att[s] = dot(u_emb, rel_feat[neigh_ir[v][s]]) over 64 dims
    #pragma unroll
    for (int s = 0; s < NS; ++s) {
        int r = neigh_ir[vv * NS + s];
        float4 rf = ldg4(rel_feat + (size_t)r * DIM + d0);
        float p = ue.x * rf.x + ue.y * rf.y + ue.z * rf.z + ue.w * rf.w;
        p += __shfl_xor(p, 1, 32);
        p += __shfl_xor(p, 2, 32);
        p += __shfl_xor(p, 4, 32);
        p += __shfl_xor(p, 8, 32);   // full 16-lane group sum (wave32 aligned)
        w_ii[s] = p;
    }
    softmax8(w_ii);

    // ---- weighted neighbor aggregation ----
    float4 au = ue;
    #pragma unroll
    for (int s = 0; s < NS; ++s) {
        int n = neigh_uu[uu * NS + s];
        fma4(au, w_uu[s], ldg4(usr_feat + (size_t)n * DIM + d0));
    }
    #pragma unroll
    for (int s = 0; s < NS; ++s) {
        int n = neigh_ui[uu * NS + s];
        fma4(au, w_ui[s], ldg4(item_feat + (size_t)n * DIM + d0));
    }
    float4 av = ve;
    #pragma unroll
    for (int s = 0; s < NS; ++s) {
        int n = neigh_iu[vv * NS + s];
        fma4(av, w_iu[s], ldg4(usr_feat + (size_t)n * DIM + d0));
    }
    #pragma unroll
    for (int s = 0; s < NS; ++s) {
        int n = neigh_ii[vv * NS + s];
        fma4(av, w_ii[s], ldg4(item_feat + (size_t)n * DIM + d0));
    }

    *(float4*)(user_x + (size_t)b * DIM + d0) = au;
    *(float4*)(item_x + (size_t)b * DIM + d0) = av;
}

// ---------------------------------------------------------------------------
// Stage 2: dual fp32 WMMA GEMM (H = relu(X @ W + b) for both sides), fused
// with the per-row dot + sigmoid head. One wave per 16-row M-tile, 8 waves
// per block, W matrices staged in LDS.
//
// V_WMMA_F32_16X16X4_F32 operand layout (wave32):
//   A (16x4):  lane l<16 holds row l, K={k0,k0+1}; lane l+16 holds K={k0+2,k0+3}
//   B (4x16):  lane l<16 holds col l, K={k0,k0+1}; lane l+16 holds K={k0+2,k0+3}
//   C/D (16x16): VGPR j, lane l -> row j + 8*(l>>4), col (l&15)
// ---------------------------------------------------------------------------
__global__ void __launch_bounds__(256) sestkgcn_gemm_head(
    const float* __restrict__ user_x, const float* __restrict__ item_x,
    const float* __restrict__ Wu, const float* __restrict__ bu,
    const float* __restrict__ Wv, const float* __restrict__ bv,
    float* __restrict__ out, int batch)
{
    __shared__ float lWu[DIM * DIM];
    __shared__ float lWv[DIM * DIM];

    // cooperative stage of both 64x64 weight matrices (16 KB each)
    for (int i = threadIdx.x; i < DIM * DIM / 4; i += 256) {
        ((float4*)lWu)[i] = ((const float4*)Wu)[i];
        ((float4*)lWv)[i] = ((const float4*)Wv)[i];
    }
    __syncthreads();

    const int wave  = threadIdx.x >> 5;        // 0..7
    const int lane  = threadIdx.x & 31;
    const int nlane = lane & 15;               // col within N-tile / row within M-tile
    const int khalf = lane >> 4;               // 0: K={0,1}, 1: K={2,3}

    const int row0 = (blockIdx.x * 8 + wave) * 16;
    if (row0 + 16 > batch) return;             // wave-uniform guard: EXEC stays all-1s
    const int mrow = row0 + nlane;

    v8f accU[4], accV[4];
    #pragma unroll
    for (int n = 0; n < 4; ++n) { accU[n] = (v8f)0.f; accV[n] = (v8f)0.f; }

    const float* urow = user_x + (size_t)mrow * DIM;
    const float* irow = item_x + (size_t)mrow * DIM;

    #pragma unroll
    for (int k0 = 0; k0 < DIM; k0 += 4) {
        const int kb = k0 + 2 * khalf;
        v2f aU = *(const v2f*)(urow + kb);
        v2f aV = *(const v2f*)(irow + kb);
        #pragma unroll
        for (int n = 0; n < 4; ++n) {
            const int col = n * 16 + nlane;
            v2f bU; bU.x = lWu[kb * DIM + col]; bU.y = lWu[(kb + 1) * DIM + col];
            v2f bV; bV.x = lWv[kb * DIM + col]; bV.y = lWv[(kb + 1) * DIM + col];
            accU[n] = __builtin_amdgcn_wmma_f32_16x16x4_f32(
                false, aU, false, bU, (short)0, accU[n], false, false);
            accV[n] = __builtin_amdgcn_wmma_f32_16x16x4_f32(
                false, aV, false, bV, (short)0, accV[n], false, false);
        }
    }

    // bias per column owned by this lane in each N-tile
    float bcu[4], bcv[4];
    #pragma unroll
    for (int n = 0; n < 4; ++n) { bcu[n] = bu[n * 16 + nlane]; bcv[n] = bv[n * 16 + nlane]; }

    // head: relu, per-row dot across 64 cols, sigmoid * 5
    float psum[8];
    #pragma unroll
    for (int j = 0; j < 8; ++j) {
        float p = 0.f;
        #pragma unroll
        for (int n = 0; n < 4; ++n) {
            float uh = fmaxf(accU[n][j] + bcu[n], 0.f);
            float vh = fmaxf(accV[n][j] + bcv[n], 0.f);
            p = fmaf(uh, vh, p);
        }
        p += __shfl_xor(p, 1, 32);
        p += __shfl_xor(p, 2, 32);
        p += __shfl_xor(p, 4, 32);
        p += __shfl_xor(p, 8, 32);
        psum[j] = p;
    }

    if (nlane == 0) {  // lanes 0 (rows 0..7) and 16 (rows 8..15)
        #pragma unroll
        for (int j = 0; j < 8; ++j) {
            float s = 5.0f / (1.0f + __expf(-psum[j]));
            out[row0 + 8 * khalf + j] = s;
        }
    }
}

extern "C" void kernel_launch(void* const* d_in, const int* in_sizes, int n_in,
                              void* d_out, int out_size, void* d_ws, size_t ws_size,
                              hipStream_t stream) {
    const int*   u          = (const int*)  d_in[0];
    const int*   v          = (const int*)  d_in[1];
    const float* usr_feat   = (const float*)d_in[2];
    const float* item_feat  = (const float*)d_in[3];
    const float* rel_feat   = (const float*)d_in[4];
    const int*   neigh_uu   = (const int*)  d_in[5];
    const float* neigh_uu_st= (const float*)d_in[6];
    const int*   neigh_ui   = (const int*)  d_in[7];
    const float* nui_rat    = (const float*)d_in[8];
    const float* nui_vot    = (const float*)d_in[9];
    const float* nui_tim    = (const float*)d_in[10];
    const int*   neigh_iu   = (const int*)  d_in[11];
    const float* niu_rat    = (const float*)d_in[12];
    const float* niu_vot    = (const float*)d_in[13];
    const float* niu_tim    = (const float*)d_in[14];
    const int*   neigh_ii   = (const int*)  d_in[15];
    const int*   neigh_ir   = (const int*)  d_in[16];
    const float* Wu         = (const float*)d_in[17];
    const float* bu         = (const float*)d_in[18];
    const float* Wv         = (const float*)d_in[19];
    const float* bv         = (const float*)d_in[20];

    const int batch = in_sizes[0];

    float* user_x = (float*)d_ws;
    float* item_x = user_x + (size_t)batch * DIM;

    const int agg_blocks = (batch * 16 + 255) / 256;
    sestkgcn_agg<<<agg_blocks, 256, 0, stream>>>(
        u, v, usr_feat, item_feat, rel_feat,
        neigh_uu, neigh_uu_st, neigh_ui, nui_rat, nui_vot, nui_tim,
        neigh_iu, niu_rat, niu_vot, niu_tim, neigh_ii, neigh_ir,
        user_x, item_x, batch);

    const int gemm_blocks = (batch + 127) / 128;  // 8 waves * 16 rows per block
    sestkgcn_gemm_head<<<gemm_blocks, 256, 0, stream>>>(
        user_x, item_x, Wu, bu, Wv, bv, (float*)d_out, batch);
}